// HistogramSelfAttention_43774306681385
// MI455X (gfx1250) — compile-verified
//
#include <hip/hip_runtime.h>

// ---------------------------------------------------------------- constants
constexpr int Bn    = 4;
constexpr int Cn    = 64;
constexpr int HALFn = 32;
constexpr int Hn    = 256;
constexpr int Wn    = 256;
constexpr int HWn   = 65536;         // H*W
constexpr int NROWS = Bn * Cn;       // 256 rows for the big v-argsort

typedef float    v2f __attribute__((ext_vector_type(2)));
typedef float    v8f __attribute__((ext_vector_type(8)));
typedef unsigned v4u __attribute__((ext_vector_type(4)));
typedef int      v8i __attribute__((ext_vector_type(8)));
typedef int      v4i __attribute__((ext_vector_type(4)));

#if defined(__AMDGCN__) && __has_builtin(__builtin_amdgcn_tensor_load_to_lds) && __has_builtin(__builtin_amdgcn_s_wait_tensorcnt)
#define HSA_TDM 1
#endif

__device__ __forceinline__ v8f wmma_f32x4(v2f a, v2f b, v8f c) {
  // D(16x16,f32) = A(16x4,f32) x B(4x16,f32) + C
  return __builtin_amdgcn_wmma_f32_16x16x4_f32(false, a, false, b, (short)0, c,
                                               false, false);
}

// ======================================================== stage 1: hist sorts
// Sort 256-length column along H for each (b, c<32, w).  Bitonic in LDS,
// (key, idx) with idx tie-break == stable argsort on random data.
__global__ __launch_bounds__(256) void hsa_sort_h(const float* __restrict__ x,
                                                  float* __restrict__ sx,
                                                  unsigned short* __restrict__ idxh) {
  __shared__ float sk[256];
  __shared__ int   si[256];
  const int bid = blockIdx.x;
  const int w = bid & 255, c = (bid >> 8) & 31, b = bid >> 13;
  const int h = threadIdx.x;
  sk[h] = x[((size_t)(b * Cn + c) * Hn + h) * Wn + w];
  si[h] = h;
  __syncthreads();
  for (int k = 2; k <= 256; k <<= 1)
    for (int j = k >> 1; j > 0; j >>= 1) {
      int p = h ^ j;
      if (p > h) {
        bool up = ((h & k) == 0);
        float a = sk[h], q = sk[p];
        int ia = si[h], ib = si[p];
        bool sw = up ? (a > q || (a == q && ia > ib))
                     : (a < q || (a == q && ia < ib));
        if (sw) { sk[h] = q; sk[p] = a; si[h] = ib; si[p] = ia; }
      }
      __syncthreads();
    }
  sx[((size_t)(b * Cn + c) * Hn + h) * Wn + w] = sk[h];
  idxh[((size_t)(b * HALFn + c) * Hn + h) * Wn + w] = (unsigned short)si[h];
}

// Sort 256-length row along W for each (b, c<32, h) of sorted_h (in-place on sx).
__global__ __launch_bounds__(256) void hsa_sort_w(float* __restrict__ sx,
                                                  unsigned short* __restrict__ idxw) {
  __shared__ float sk[256];
  __shared__ int   si[256];
  const int bid = blockIdx.x;
  const int h = bid & 255, c = (bid >> 8) & 31, b = bid >> 13;
  const int w = threadIdx.x;
  const size_t rowb = ((size_t)(b * Cn + c) * Hn + h) * Wn;
  sk[w] = sx[rowb + w];
  si[w] = w;
  __syncthreads();
  for (int k = 2; k <= 256; k <<= 1)
    for (int j = k >> 1; j > 0; j >>= 1) {
      int p = w ^ j;
      if (p > w) {
        bool up = ((w & k) == 0);
        float a = sk[w], q = sk[p];
        int ia = si[w], ib = si[p];
        bool sw = up ? (a > q || (a == q && ia > ib))
                     : (a < q || (a == q && ia < ib));
        if (sw) { sk[w] = q; sk[p] = a; si[w] = ib; si[p] = ia; }
      }
      __syncthreads();
    }
  sx[rowb + w] = sk[w];
  idxw[((size_t)(b * HALFn + c) * Hn + h) * Wn + w] = (unsigned short)si[w];
}

// Copy upper 32 channels of x into x_sorted buffer.
__global__ __launch_bounds__(256) void hsa_copy_upper(const float* __restrict__ x,
                                                      float* __restrict__ sx) {
  size_t g = (size_t)blockIdx.x * 256 + threadIdx.x;  // B*32*HW total
  size_t n = g & (HWn - 1);
  size_t r = g >> 16;
  int c = (int)(r & 31), b = (int)(r >> 5);
  size_t o = ((size_t)(b * Cn + HALFn + c)) * HWn + n;
  sx[o] = x[o];
}

// ======================================================== WMMA f32 GEMM
// Y[b](Orows x HW) = Wm(Orows x 64) @ X[b](64 x HW).  Block tile 64(o) x 128(n),
// 8 waves, each wave 16x64 (4 accumulators), K=64 resident in LDS.
// B-tile (64x128 of X) streamed with the Tensor Data Mover when available.
__global__ __launch_bounds__(256) void hsa_gemm(const float* __restrict__ Wm,
                                                const float* __restrict__ X,
                                                float* __restrict__ Y, int Orows) {
  __shared__ float Xs[64 * 128];   // [k][n], packed (TDM writes contiguously)
  __shared__ float Ws[64 * 65];    // [k][o], padded
  const int tid = threadIdx.x;
  const int n0 = blockIdx.x * 128;
  const int o0 = blockIdx.y * 64;
  const int b  = blockIdx.z;
  const float* Xb = X + (size_t)b * 64 * HWn;
#ifdef HSA_TDM
  if (tid < 32) {
    unsigned long long ga = (unsigned long long)(const void*)(Xb + n0);
    unsigned ldsa = (unsigned)(unsigned long long)(void*)&Xs[0];
    v4u g0;
    g0[0] = 1u;                                            // count = 1
    g0[1] = ldsa;                                          // lds_addr
    g0[2] = (unsigned)(ga & 0xFFFFFFFFull);                // global_addr lo
    g0[3] = (unsigned)((ga >> 32) & 0x1FFFFFFull) | (2u << 30);  // hi + type=2
    v8i g1;
    g1[0] = (int)(2u << 16);                               // data_size = 4B
    g1[1] = (int)(((unsigned)HWn & 0xFFFFu) << 16);        // tensor_dim0 lo16
    g1[2] = (int)(((unsigned)HWn >> 16) | (64u << 16));    // dim0 hi | dim1 lo
    g1[3] = (int)(128u << 16);                             // dim1 hi | tile_dim0=128
    g1[4] = 64;                                            // tile_dim1=64, tile_dim2=0
    g1[5] = (int)HWn;                                      // dim0_stride lo32
    g1[6] = 0;                                             // stride hi / dim1_stride
    g1[7] = 0;
    v4i zz4 = {0, 0, 0, 0};
    v8i zz8 = {0, 0, 0, 0, 0, 0, 0, 0};
    __builtin_amdgcn_tensor_load_to_lds(g0, g1, zz4, zz4, zz8, 0);
  }
#else
  for (int i = tid; i < 64 * 128; i += 256) {
    int c = i >> 7, n = i & 127;
    Xs[i] = Xb[(size_t)c * HWn + n0 + n];
  }
#endif
  for (int i = tid; i < 64 * 64; i += 256) {
    int o = i >> 6, c = i & 63;
    Ws[c * 65 + o] = Wm[(size_t)(o0 + o) * 64 + c];
  }
#ifdef HSA_TDM
  __builtin_amdgcn_s_wait_tensorcnt(0);
#endif
  __syncthreads();
  const int lane = tid & 31, wave = tid >> 5;
  const int wo = wave >> 1, wn = (wave & 1) * 64;
  const int m = lane & 15, hi = lane >> 4;
  v8f acc[4] = {};
  for (int kk = 0; kk < 16; ++kk) {
    const int kb = kk * 4;
    v2f a;
    a.x = Ws[(kb + 2 * hi) * 65 + wo * 16 + m];
    a.y = Ws[(kb + 2 * hi + 1) * 65 + wo * 16 + m];
#pragma unroll
    for (int t = 0; t < 4; ++t) {
      v2f bb;
      bb.x = Xs[(kb + hi) * 128 + wn + t * 16 + m];
      bb.y = Xs[(kb + 2 + hi) * 128 + wn + t * 16 + m];
      acc[t] = wmma_f32x4(a, bb, acc[t]);
    }
  }
  float* Yb = Y + (size_t)b * Orows * HWn;
  for (int t = 0; t < 4; ++t)
    for (int r = 0; r < 8; ++r)
      Yb[(size_t)(o0 + wo * 16 + r + 8 * hi) * HWn + (n0 + wn + t * 16 + m)] =
          acc[t][r];
}

// ======================================================== depthwise conv 3x3
__global__ __launch_bounds__(256) void hsa_dwconv(const float* __restrict__ Yin,
                                                  const float* __restrict__ Wdw,
                                                  float* __restrict__ Z) {
  __shared__ float t[18][18];
  const int bo = blockIdx.z;           // b*320 + o
  const int o = bo % 320;
  const int h0 = blockIdx.y * 16, w0 = blockIdx.x * 16;
  const float* p = Yin + (size_t)bo * HWn;
  const int tx = threadIdx.x & 15, ty = threadIdx.x >> 4;
  for (int i = threadIdx.x; i < 18 * 18; i += 256) {
    int yy = i / 18 - 1 + h0, xx = i % 18 - 1 + w0;
    float v = 0.f;
    if (yy >= 0 && yy < Hn && xx >= 0 && xx < Wn) v = p[yy * Wn + xx];
    t[i / 18][i % 18] = v;
  }
  __syncthreads();
  float acc = 0.f;
#pragma unroll
  for (int dy = 0; dy < 3; ++dy)
#pragma unroll
    for (int dx = 0; dx < 3; ++dx)
      acc += t[ty + dy][tx + dx] * Wdw[o * 9 + dy * 3 + dx];
  Z[(size_t)bo * HWn + (h0 + ty) * Wn + (w0 + tx)] = acc;
}

// ======================================================== radix argsort of v
__device__ __forceinline__ unsigned float_key(float f) {
  unsigned u = __float_as_uint(f);
  return (u & 0x80000000u) ? ~u : (u | 0x80000000u);
}

__global__ __launch_bounds__(256) void hsa_radix_init(const float* __restrict__ Z,
                                                      unsigned* __restrict__ keys,
                                                      unsigned* __restrict__ idx) {
  size_t g = (size_t)blockIdx.x * 256 + threadIdx.x;  // NROWS*HW total
  unsigned n = (unsigned)(g & (HWn - 1));
  size_t r = g >> 16;
  size_t src = ((size_t)((r >> 6) * 320 + 256 + (r & 63))) * HWn + n; // v chunk
  keys[g] = float_key(Z[src]);
  idx[g] = n;
}

__global__ __launch_bounds__(256) void hsa_radix_hist(const unsigned* __restrict__ keys,
                                                      unsigned* __restrict__ hist,
                                                      int shift) {
  __shared__ unsigned hcnt[256];
  hcnt[threadIdx.x] = 0;
  __syncthreads();
  const int blk = blockIdx.x;          // r*256 + chunk
  const int r = blk >> 8, ch = blk & 255;
  unsigned key = keys[((size_t)r << 16) + ch * 256 + threadIdx.x];
  atomicAdd(&hcnt[(key >> shift) & 255], 1u);
  __syncthreads();
  hist[((size_t)r << 16) | ((unsigned)threadIdx.x << 8) | (unsigned)ch] =
      hcnt[threadIdx.x];               // layout [row][digit][chunk]
}

__global__ __launch_bounds__(256) void hsa_radix_scan(unsigned* __restrict__ hist) {
  __shared__ unsigned s[256];
  unsigned* base = hist + ((size_t)blockIdx.x << 16);
  const int t0 = threadIdx.x * 256;
  unsigned sum = 0;
  for (int i = 0; i < 256; ++i) sum += base[t0 + i];
  s[threadIdx.x] = sum;
  __syncthreads();
  for (int off = 1; off < 256; off <<= 1) {
    unsigned u = (threadIdx.x >= off) ? s[threadIdx.x - off] : 0u;
    __syncthreads();
    s[threadIdx.x] += u;
    __syncthreads();
  }
  unsigned run = (threadIdx.x == 0) ? 0u : s[threadIdx.x - 1];
  for (int i = 0; i < 256; ++i) { unsigned c = base[t0 + i]; base[t0 + i] = run; run += c; }
}

__global__ __launch_bounds__(256) void hsa_radix_scatter(
    const unsigned* __restrict__ keys, const unsigned* __restrict__ idx,
    unsigned* __restrict__ keys_o, unsigned* __restrict__ idx_o,
    const unsigned* __restrict__ scan, int shift) {
  __shared__ unsigned short dg[256];
  const int blk = blockIdx.x;
  const int r = blk >> 8, ch = blk & 255;
  const size_t bi = ((size_t)r << 16) + ch * 256 + threadIdx.x;
  unsigned key = keys[bi], id = idx[bi];
  unsigned d = (key >> shift) & 255;
  dg[threadIdx.x] = (unsigned short)d;
  __syncthreads();
  unsigned rank = 0;
  for (int j = 0; j < threadIdx.x; ++j) rank += (dg[j] == (unsigned short)d);
  unsigned pos = scan[((size_t)r << 16) | (d << 8) | (unsigned)ch] + rank;
  keys_o[((size_t)r << 16) + pos] = key;
  idx_o[((size_t)r << 16) + pos] = id;
}

// ======================================================== gather q1,k1,q2,k2,v
__global__ __launch_bounds__(256) void hsa_gather(const float* __restrict__ Z,
                                                  const unsigned* __restrict__ idx,
                                                  float* __restrict__ G) {
  size_t g = (size_t)blockIdx.x * 256 + threadIdx.x;  // B*5*64*HW total
  unsigned n = (unsigned)(g & (HWn - 1));
  size_t ch = g >> 16;                 // flat channel b*320 + q*64 + c
  int c = (int)(ch & 63);
  int b = (int)(ch / 320);
  unsigned src = idx[((size_t)(b * 64 + c) << 16) + n];
  G[g] = Z[(ch << 16) + src];
}

// ======================================================== L2-norm scales
__global__ __launch_bounds__(256) void hsa_norms(const float* __restrict__ G,
                                                 float* __restrict__ scales) {
  __shared__ float red[256];
  const int blk = blockIdx.x;          // t*2048 + b*512 + h*64 + i
  const int i = blk & 63, h = (blk >> 6) & 7, b = (blk >> 9) & 3, t = blk >> 11;
  const int cp = i >> 3, f = i & 7;
  const bool pb = (t < 2);             // t: 0=q1,1=k1 (branch p), 2=q2,3=k2 (c)
  const float* src = G + ((size_t)((b * 5 + t) * 64) + h * 8 + cp) * HWn;
  float s = 0.f;
  for (int g = threadIdx.x; g < 8192; g += 256) {
    float v = pb ? src[(size_t)f * 8192 + g] : src[(size_t)g * 8 + f];
    s += v * v;
  }
  red[threadIdx.x] = s;
  __syncthreads();
  for (int o = 128; o > 0; o >>= 1) {
    if (threadIdx.x < o) red[threadIdx.x] += red[threadIdx.x + o];
    __syncthreads();
  }
  if (threadIdx.x == 0) scales[blk] = 1.f / fmaxf(sqrtf(red[0]), 1e-12f);
}

// ======================================================== QK^T + softmax (WMMA)
__global__ __launch_bounds__(256) void hsa_attn_qk(const float* __restrict__ G,
                                                   const float* __restrict__ scales,
                                                   const float* __restrict__ temp,
                                                   float* __restrict__ attn) {
  __shared__ float Qs[64][65];
  __shared__ float Ks[64][65];
  __shared__ float As[64][65];
  const int blk = blockIdx.x;          // br*32 + b*8 + h
  const int h = blk & 7, b = (blk >> 3) & 3, br = blk >> 5;
  const int tid = threadIdx.x, lane = tid & 31, wave = tid >> 5;
  const int ti = wave >> 1, tj = (wave & 1) * 2;
  const int m = lane & 15, hi = lane >> 4;
  const float* Qsrc = G + ((size_t)((b * 5 + br * 2) * 64) + h * 8) * HWn;
  const float* Ksrc = G + ((size_t)((b * 5 + br * 2 + 1) * 64) + h * 8) * HWn;
  const float* qsc = scales + (size_t)(br * 2) * 2048 + b * 512 + h * 64;
  const float* ksc = scales + (size_t)(br * 2 + 1) * 2048 + b * 512 + h * 64;
  v8f acc0 = {}, acc1 = {};
  for (int g0 = 0; g0 < 8192; g0 += 64) {
    __syncthreads();
    for (int l = tid; l < 4096; l += 256) {
      int i = l >> 6, gg = l & 63;
      int cp = i >> 3, f = i & 7;
      size_t off = (size_t)cp * HWn + (br == 0 ? (size_t)f * 8192 + g0 + gg
                                               : (size_t)(g0 + gg) * 8 + f);
      Qs[i][gg] = Qsrc[off] * qsc[i];
      Ks[i][gg] = Ksrc[off] * ksc[i];
    }
    __syncthreads();
    for (int kk = 0; kk < 16; ++kk) {
      const int kb = kk * 4;
      v2f a;
      a.x = Qs[ti * 16 + m][kb + 2 * hi];
      a.y = Qs[ti * 16 + m][kb + 2 * hi + 1];
      v2f b0, b1;
      b0.x = Ks[tj * 16 + m][kb + hi];
      b0.y = Ks[tj * 16 + m][kb + 2 + hi];
      acc0 = wmma_f32x4(a, b0, acc0);
      b1.x = Ks[(tj + 1) * 16 + m][kb + hi];
      b1.y = Ks[(tj + 1) * 16 + m][kb + 2 + hi];
      acc1 = wmma_f32x4(a, b1, acc1);
    }
  }
  __syncthreads();
  const float tv = temp[h];
#pragma unroll
  for (int r = 0; r < 8; ++r) {
    As[ti * 16 + r + 8 * hi][tj * 16 + m] = acc0[r] * tv;
    As[ti * 16 + r + 8 * hi][(tj + 1) * 16 + m] = acc1[r] * tv;
  }
  __syncthreads();
  if (tid < 64) {
    float mx = -3.4e38f;
    for (int j = 0; j < 64; ++j) mx = fmaxf(mx, As[tid][j]);
    float sm = 0.f;
    for (int j = 0; j < 64; ++j) sm += __expf(As[tid][j] - mx);
    const float inv = 1.f / sm;
    float* dst = attn + ((size_t)blk << 12) + tid * 64;
    for (int j = 0; j < 64; ++j) dst[j] = __expf(As[tid][j] - mx) * inv;
  }
}

// ======================================================== attn @ V (WMMA)
// Output mapping is factor-first for BOTH branches (reference's final reshape);
// only the V gathering differs.  Branch c multiplies into O1 (out_p * out_c).
template <bool PB>
__global__ __launch_bounds__(256) void hsa_attn_v(const float* __restrict__ G,
                                                  const float* __restrict__ attn,
                                                  float* __restrict__ O1) {
  __shared__ float As[64][65];     // [j][i]
  __shared__ float Vs[64 * 128];   // [j][g]
  const int b = blockIdx.z, h = blockIdx.y, g0 = blockIdx.x * 128;
  const int br = PB ? 0 : 1;
  const int tid = threadIdx.x;
  const float* at = attn + ((size_t)(br * 32 + b * 8 + h) << 12);
  for (int l = tid; l < 4096; l += 256) {
    int i = l >> 6, j = l & 63;
    As[j][i] = at[l];
  }
  const float* Vsrc = G + ((size_t)((b * 5 + 4) * 64) + h * 8) * HWn;
  for (int l = tid; l < 64 * 128; l += 256) {
    int j = l >> 7, gg = l & 127;
    int cp = j >> 3, f = j & 7;
    size_t off = (size_t)cp * HWn + (PB ? (size_t)f * 8192 + g0 + gg
                                        : (size_t)(g0 + gg) * 8 + f);
    Vs[l] = Vsrc[off];
  }
  __syncthreads();
  const int lane = tid & 31, wave = tid >> 5;
  const int wo = wave >> 1, wn = (wave & 1) * 64;
  const int m = lane & 15, hi = lane >> 4;
  v8f acc[4] = {};
  for (int kk = 0; kk < 16; ++kk) {
    const int kb = kk * 4;
    v2f a;
    a.x = As[kb + 2 * hi][wo * 16 + m];
    a.y = As[kb + 2 * hi + 1][wo * 16 + m];
#pragma unroll
    for (int t = 0; t < 4; ++t) {
      v2f bb;
      bb.x = Vs[(kb + hi) * 128 + wn + t * 16 + m];
      bb.y = Vs[(kb + 2 + hi) * 128 + wn + t * 16 + m];
      acc[t] = wmma_f32x4(a, bb, acc[t]);
    }
  }
  float* Ob = O1 + ((size_t)(b * 64) + h * 8) * HWn;
  for (int t = 0; t < 4; ++t)
    for (int r = 0; r < 8; ++r) {
      int i = wo * 16 + r + 8 * hi;
      int g = g0 + wn + t * 16 + m;
      size_t off = (size_t)(i >> 3) * HWn + (size_t)(i & 7) * 8192 + g;
      if (PB) Ob[off] = acc[t][r];
      else    Ob[off] *= acc[t][r];
    }
}

// ======================================================== inverse perms + final
__global__ __launch_bounds__(256) void hsa_unsort_v(const float* __restrict__ O1,
                                                    const unsigned* __restrict__ idx,
                                                    float* __restrict__ U) {
  size_t g = (size_t)blockIdx.x * 256 + threadIdx.x;  // B*64*HW
  unsigned n = (unsigned)(g & (HWn - 1));
  size_t rc = g >> 16;
  U[(rc << 16) + idx[(rc << 16) + n]] = O1[g];
}

__global__ __launch_bounds__(256) void hsa_finalize_w(const float* __restrict__ P,
                                                      const unsigned short* __restrict__ idxw,
                                                      float* __restrict__ T1,
                                                      float* __restrict__ Out) {
  size_t g = (size_t)blockIdx.x * 256 + threadIdx.x;  // B*64*HW
  int w = (int)(g & 255), h = (int)((g >> 8) & 255);
  int c = (int)((g >> 16) & 63), b = (int)(g >> 22);
  float v = P[g];
  if (c >= HALFn) {
    Out[g] = v;
  } else {
    size_t rb = ((size_t)(b * HALFn + c) * Hn + h) * Wn;
    T1[rb + idxw[rb + w]] = v;     // scatter = gather by argsort(idx_w)
  }
}

__global__ __launch_bounds__(256) void hsa_finalize_h(const float* __restrict__ T1,
                                                      const unsigned short* __restrict__ idxh,
                                                      float* __restrict__ Out) {
  size_t g = (size_t)blockIdx.x * 256 + threadIdx.x;  // B*32*HW
  int w = (int)(g & 255), h = (int)((g >> 8) & 255);
  int c = (int)((g >> 16) & 31), b = (int)(g >> 21);
  size_t rb = ((size_t)(b * HALFn + c) * Hn + h) * Wn;
  int ih = idxh[rb + w];
  Out[((size_t)(b * Cn + c) * Hn + ih) * Wn + w] = T1[rb + w];
}

// ======================================================== host launcher
extern "C" void kernel_launch(void* const* d_in, const int* in_sizes, int n_in,
                              void* d_out, int out_size, void* d_ws, size_t ws_size,
                              hipStream_t stream) {
  (void)in_sizes; (void)n_in; (void)out_size; (void)ws_size;
  const float* x      = (const float*)d_in[0];
  const float* w_qkv  = (const float*)d_in[1];
  const float* w_dw   = (const float*)d_in[2];
  const float* w_proj = (const float*)d_in[3];
  const float* temp   = (const float*)d_in[4];
  float* out = (float*)d_out;
  char* ws = (char*)d_ws;
  const size_t MB = 1ull << 20;

  float*    sx    = (float*)(ws + 0);          //  64 MB, dead after qkv GEMM
  float*    ybuf  = (float*)(ws + 64 * MB);    // 320 MB, reused: radix temps, then G
  float*    zbuf  = (float*)(ws + 384 * MB);   // 320 MB, reused after gather: U/P/T1
  unsigned* keysA = (unsigned*)(ws + 64 * MB);
  unsigned* keysB = (unsigned*)(ws + 128 * MB);
  unsigned* idxB  = (unsigned*)(ws + 192 * MB);
  unsigned* hist  = (unsigned*)(ws + 256 * MB);
  unsigned* idxA  = (unsigned*)(ws + 704 * MB);  // 64 MB, kept until unsort
  unsigned short* idxh = (unsigned short*)(ws + 768 * MB);  // 16 MB
  unsigned short* idxw = (unsigned short*)(ws + 784 * MB);  // 16 MB
  float*    attnb  = (float*)(ws + 800 * MB);  // 1 MB
  float*    scales = (float*)(ws + 802 * MB);  // 32 KB
  float*    O1   = (float*)(ws + 0);           // reuse sx
  float*    Ubuf = (float*)(ws + 384 * MB);    // reuse z
  float*    Pbuf = (float*)(ws + 448 * MB);
  float*    T1   = (float*)(ws + 512 * MB);
  float*    G    = ybuf;

  // 1-2) histogram sorts + copy of untouched channels
  hsa_sort_h<<<Bn * HALFn * Wn, 256, 0, stream>>>(x, sx, idxh);
  hsa_sort_w<<<Bn * HALFn * Hn, 256, 0, stream>>>(sx, idxw);
  hsa_copy_upper<<<(Bn * HALFn * HWn) / 256, 256, 0, stream>>>(x, sx);
  // 3) y = w_qkv @ x_sorted   (WMMA + TDM)
  hsa_gemm<<<dim3(HWn / 128, 5, Bn), 256, 0, stream>>>(w_qkv, sx, ybuf, 320);
  // 4) depthwise 3x3
  hsa_dwconv<<<dim3(Wn / 16, Hn / 16, Bn * 320), 256, 0, stream>>>(ybuf, w_dw, zbuf);
  // 5) stable LSD radix argsort of v (4x8-bit passes)
  hsa_radix_init<<<(NROWS * HWn) / 256, 256, 0, stream>>>(zbuf, keysA, idxA);
  unsigned *ki = keysA, *ii = idxA, *ko = keysB, *io = idxB;
  for (int p = 0; p < 4; ++p) {
    hsa_radix_hist<<<NROWS * 256, 256, 0, stream>>>(ki, hist, 8 * p);
    hsa_radix_scan<<<NROWS, 256, 0, stream>>>(hist);
    hsa_radix_scatter<<<NROWS * 256, 256, 0, stream>>>(ki, ii, ko, io, hist, 8 * p);
    unsigned* t;
    t = ki; ki = ko; ko = t;
    t = ii; ii = io; io = t;
  }
  // ii == idxA after 4 passes
  // 6) gather q1,k1,q2,k2,v into sorted domain
  hsa_gather<<<(Bn * 5 * 64 * HWn) / 256, 256, 0, stream>>>(zbuf, ii, G);
  // 7) attention (both branches)
  hsa_norms<<<4 * Bn * 8 * 64, 256, 0, stream>>>(G, scales);
  hsa_attn_qk<<<2 * Bn * 8, 256, 0, stream>>>(G, scales, temp, attnb);
  hsa_attn_v<true><<<dim3(8192 / 128, 8, Bn), 256, 0, stream>>>(G, attnb, O1);
  hsa_attn_v<false><<<dim3(8192 / 128, 8, Bn), 256, 0, stream>>>(G, attnb, O1);
  // 8) inverse v-permutation, projection, inverse histogram sorts
  hsa_unsort_v<<<(Bn * Cn * HWn) / 256, 256, 0, stream>>>(O1, ii, Ubuf);
  hsa_gemm<<<dim3(HWn / 128, 1, Bn), 256, 0, stream>>>(w_proj, Ubuf, Pbuf, 64);
  hsa_finalize_w<<<(Bn * Cn * HWn) / 256, 256, 0, stream>>>(Pbuf, idxw, T1, out);
  hsa_finalize_h<<<(Bn * HALFn * HWn) / 256, 256, 0, stream>>>(T1, idxh, out);
}